// RelationLayer_87531433492663
// MI455X (gfx1250) — compile-verified
//
#include <hip/hip_runtime.h>

typedef __attribute__((ext_vector_type(2))) float v2f;
typedef __attribute__((ext_vector_type(8))) float v8f;

#define BSZ 4
#define SSZ 512
#define HSZ 128
#define TI  4   // i-rows per block in pairwise kernel

// -------------------------------------------------------------------------
// Kernel 1: G0 = inp @ W0^T + b0 ; G1 = inp @ W1^T + b1
// M = B*S = 2048, N = H = 128, K = H = 128.
// One wave computes one 16x16 output tile for BOTH G0 and G1 (shared A).
// Uses V_WMMA_F32_16X16X4_F32, K-loop of 32 steps.
//
// f32 A-tile (16x4) layout: lanes 0-15 hold rows M=0..15 with K={k0,k0+1},
// lanes 16-31 hold the same rows with K={k0+2,k0+3}.
// f32 B-tile (4x16) layout mirrors it: lane&15 = N, lane>>4 selects K-pair.
// Since Bmat[k][n] = W[n][k] (row-major W, out = inp@W^T), the B-tile load is
// reading 2 consecutive-k floats from row n of W — identical access shape to A.
// C/D (16x16 f32, 8 VGPRs): VGPR v, lanes 0-15 -> M=v, lanes 16-31 -> M=v+8,
// N = lane&15.
// -------------------------------------------------------------------------
__global__ __launch_bounds__(128) void gates_wmma_kernel(
    const float* __restrict__ inp,
    const float* __restrict__ W0, const float* __restrict__ b0,
    const float* __restrict__ W1, const float* __restrict__ b1,
    float* __restrict__ G0, float* __restrict__ G1)
{
  const int lane  = threadIdx.x & 31;
  const int wave  = threadIdx.x >> 5;
  const int tile  = blockIdx.x * 4 + wave;   // 0..1023
  const int ntile = tile & 7;                // 128/16 = 8 N-tiles
  const int mtile = tile >> 3;               // 2048/16 = 128 M-tiles

  const int r  = lane & 15;                  // row within tile (A: M, B: N)
  const int kb = (lane >> 4) * 2;            // K sub-offset: 0 or 2

  const float* Aptr  = inp + (size_t)(mtile * 16 + r) * HSZ + kb;
  const float* B0ptr = W0  + (size_t)(ntile * 16 + r) * HSZ + kb;
  const float* B1ptr = W1  + (size_t)(ntile * 16 + r) * HSZ + kb;

  v8f acc0 = {};
  v8f acc1 = {};
  #pragma unroll 4
  for (int k0 = 0; k0 < HSZ; k0 += 4) {
    v2f a  = *(const v2f*)(Aptr  + k0);
    v2f w0 = *(const v2f*)(B0ptr + k0);
    v2f w1 = *(const v2f*)(B1ptr + k0);
    acc0 = __builtin_amdgcn_wmma_f32_16x16x4_f32(
        /*neg_a=*/false, a, /*neg_b=*/false, w0, (short)0, acc0, false, false);
    acc1 = __builtin_amdgcn_wmma_f32_16x16x4_f32(
        /*neg_a=*/false, a, /*neg_b=*/false, w1, (short)0, acc1, false, false);
  }

  const int col  = ntile * 16 + (lane & 15);
  const int mofs = (lane >> 4) * 8;
  const float bias0 = b0[col];
  const float bias1 = b1[col];
  #pragma unroll
  for (int v = 0; v < 8; ++v) {
    const size_t idx = (size_t)(mtile * 16 + mofs + v) * HSZ + col;
    G0[idx] = acc0[v] + bias0;
    G1[idx] = acc1[v] + bias1;
  }
}

// -------------------------------------------------------------------------
// Kernel 2: out[b,i,h] = tanh( m_i/len_b * sum_j m_j*inp[b,j,h]*sigmoid(g0_ih+g1_jh) )
// Block = (b, 4 consecutive i), 128 threads (thread = h). The j-loop loads
// inp[b,j,h] and G1[b,j,h] once and feeds 4 sigmoids (one per i) — exp-bound.
// -------------------------------------------------------------------------
__global__ __launch_bounds__(128) void pairwise_kernel(
    const float* __restrict__ inp,
    const int*   __restrict__ masks,
    const float* __restrict__ G0,
    const float* __restrict__ G1,
    float* __restrict__ out)
{
  const int h     = threadIdx.x;                 // 0..127
  const int b     = blockIdx.x / (SSZ / TI);
  const int ibase = (blockIdx.x % (SSZ / TI)) * TI;

  float g0[TI];
  float acc[TI];
  #pragma unroll
  for (int t = 0; t < TI; ++t) {
    g0[t]  = G0[((size_t)(b * SSZ + ibase + t) * HSZ) + h];
    acc[t] = 0.0f;
  }

  const float* inpB = inp + (size_t)b * SSZ * HSZ;
  const float* g1B  = G1  + (size_t)b * SSZ * HSZ;
  const int*   mB   = masks + b * SSZ;

  int len = 0;
  for (int j = 0; j < SSZ; ++j) {
    const int mj = mB[j];            // block-uniform -> scalar branch
    len += mj;
    if (mj) {
      const float x  = inpB[(size_t)j * HSZ + h];
      const float g1 = g1B [(size_t)j * HSZ + h];
      #pragma unroll
      for (int t = 0; t < TI; ++t) {
        const float s = 1.0f / (1.0f + __expf(-(g0[t] + g1)));
        acc[t] += x * s;
      }
    }
  }

  const float inv_len = 1.0f / (float)len;
  #pragma unroll
  for (int t = 0; t < TI; ++t) {
    const float mi = (float)mB[ibase + t];
    out[((size_t)(b * SSZ + ibase + t) * HSZ) + h] = tanhf(mi * acc[t] * inv_len);
  }
}

extern "C" void kernel_launch(void* const* d_in, const int* in_sizes, int n_in,
                              void* d_out, int out_size, void* d_ws, size_t ws_size,
                              hipStream_t stream) {
  const float* inp   = (const float*)d_in[0];
  const int*   masks = (const int*)  d_in[1];
  const float* W0    = (const float*)d_in[2];
  const float* b0    = (const float*)d_in[3];
  const float* W1    = (const float*)d_in[4];
  const float* b1    = (const float*)d_in[5];
  float* out = (float*)d_out;

  float* G0 = (float*)d_ws;                          // 2048*128 f32 = 1 MB
  float* G1 = G0 + (size_t)BSZ * SSZ * HSZ;          // 1 MB

  // 1024 wave-tiles (128 M-tiles x 8 N-tiles), 4 waves per block.
  gates_wmma_kernel<<<256, 128, 0, stream>>>(inp, W0, b0, W1, b1, G0, G1);
  // 2048 (b,i)-rows / TI per block.
  pairwise_kernel<<<(BSZ * SSZ) / TI, 128, 0, stream>>>(inp, masks, G0, G1, out);
}